// TH_LSTM_57
// MI455X (gfx1250) — compile-verified
//
#include <hip/hip_runtime.h>

#define BATCH 128
#define LSEQ  55
#define KWIN  8
#define PSZ   4
#define DDIM  64
#define UNITS 48
#define MROWS 49
#define NW    32      // waves per block = K*PS groups
#define MSPAD 33      // memsum row stride (bank-conflict pad)

typedef float v2f __attribute__((ext_vector_type(2)));
typedef float v8f __attribute__((ext_vector_type(8)));

enum {G_SH=0,G_SX,G_FH,G_FX,G_FS,G_IH,G_IX,G_IS,G_TH,G_TX,G_TS,
      G_ELPH,G_ELPX,G_ELPS,G_OH,G_OX,G_OS,G_AQ,G_AH,G_V,G_H,G_E,G_G,NGATES};

struct Params {
  const float* w[NGATES];
  const float* bw[NGATES];
  const float* w_st;  const float* bw_st;
  const float* b_s;   const float* b_f;  const float* b_i; const float* b_T;
  const float* b_o;   const float* b_a;  const float* b_h;
  const float* b_elp;
  const float* gamma; const float* beta;
};

__device__ __forceinline__ float wsum(float v) {
  #pragma unroll
  for (int off = 16; off > 0; off >>= 1) v += __shfl_xor(v, off, 32);
  return v;
}
__device__ __forceinline__ float wmaxr(float v) {
  #pragma unroll
  for (int off = 16; off > 0; off >>= 1) v = fmaxf(v, __shfl_xor(v, off, 32));
  return v;
}

// Hardware TANH transcendental (CDNA5 TRANS-class op). Prefer the builtin so
// the compiler handles the TRANS hazard; otherwise inline v_tanh_f32 with an
// embedded v_nop (ISA: 1 independent op required after a TRANS before use).
__device__ __forceinline__ float fast_tanh(float x) {
#if __has_builtin(__builtin_amdgcn_tanhf)
  return __builtin_amdgcn_tanhf(x);
#else
  float r;
  asm volatile("v_tanh_f32 %0, %1\n\tv_nop" : "=v"(r) : "v"(x));
  return r;
#endif
}
// sigmoid(x) == 0.5*tanh(x/2) + 0.5  -> single TRANS op instead of exp+rcp
__device__ __forceinline__ float sigm(float x) {
  return fmaf(0.5f, fast_tanh(0.5f * x), 0.5f);
}

__global__ __launch_bounds__(1024, 1)
void th_lstm_kernel(const float* __restrict__ data, Params P, float* __restrict__ out)
{
  const int b    = blockIdx.x;
  const int tid  = threadIdx.x;
  const int wid  = tid >> 5;
  const int lane = tid & 31;
  const int kk   = wid >> 2;
  const int ps   = wid & 3;
  const int d0   = lane, d1 = lane + 32;
  const int wi0  = wid * DDIM + d0, wi1 = wid * DDIM + d1;
  const int pi0  = ps  * DDIM + d0, pi1 = ps  * DDIM + d1;

  __shared__ float rowsumS[LSEQ * PSZ];   // sum_{f=1..63} data[b,l,ps,f]
  __shared__ float t0S[LSEQ * PSZ];       // data[b,l,ps,0]
  __shared__ float memsumS[MROWS * MSPAD];
  __shared__ float AbufS[NW * DDIM];      // per-step attention bias vector
  __shared__ float wahS[NW * DDIM];       // cached w_ah
  __shared__ float partS[NW * UNITS];     // per-wave score partials
  __shared__ float scS[UNITS];
  __shared__ float alphaS[UNITS];

  for (int i = tid; i < MROWS * MSPAD; i += 1024) memsumS[i] = 0.f;

  const float* dbase = data + (size_t)b * LSEQ * PSZ * DDIM;
  __builtin_prefetch(dbase + tid * 8, 0, 0);   // global_prefetch_b8

  // Row sums of the input (xsum components and timestamps), wave-parallel.
  for (int pr = wid; pr < LSEQ * PSZ; pr += NW) {
    const float* rp = dbase + pr * DDIM;
    float v0 = rp[lane], v1 = rp[lane + 32];
    float tot = wsum(v0 + v1);
    if (lane == 0) { rowsumS[pr] = tot - v0; t0S[pr] = v0; }
  }

  wahS[wi0] = P.w[G_AH][wi0];
  wahS[wi1] = P.w[G_AH][wi1];
  const float wv = wsum(P.w[G_V][wi0] + P.w[G_V][wi1]);  // sum_d w_v[k,ps,:]

  // ---- cache per-lane weights (fixed (k,ps,d0/d1) for whole scan) ----
  const float wsh0=P.w[G_SH][wi0],  wsh1=P.w[G_SH][wi1];
  const float wsx0=P.w[G_SX][wi0],  wsx1=P.w[G_SX][wi1];
  const float wfh0=P.w[G_FH][wi0],  wfh1=P.w[G_FH][wi1];
  const float wfx0=P.w[G_FX][wi0],  wfx1=P.w[G_FX][wi1];
  const float wfs0=P.w[G_FS][wi0],  wfs1=P.w[G_FS][wi1];
  const float wih0=P.w[G_IH][wi0],  wih1=P.w[G_IH][wi1];
  const float wix0=P.w[G_IX][wi0],  wix1=P.w[G_IX][wi1];
  const float wis0=P.w[G_IS][wi0],  wis1=P.w[G_IS][wi1];
  const float wTh0=P.w[G_TH][wi0],  wTh1=P.w[G_TH][wi1];
  const float wTx0=P.w[G_TX][wi0],  wTx1=P.w[G_TX][wi1];
  const float wTs0=P.w[G_TS][wi0],  wTs1=P.w[G_TS][wi1];
  const float weh0=P.w[G_ELPH][wi0],weh1=P.w[G_ELPH][wi1];
  const float wex0=P.w[G_ELPX][wi0],wex1=P.w[G_ELPX][wi1];
  const float wes0=P.w[G_ELPS][wi0],wes1=P.w[G_ELPS][wi1];
  const float woh0=P.w[G_OH][wi0],  woh1=P.w[G_OH][wi1];
  const float wox0=P.w[G_OX][wi0],  wox1=P.w[G_OX][wi1];
  const float wos0=P.w[G_OS][wi0],  wos1=P.w[G_OS][wi1];
  const float waq0=P.w[G_AQ][wi0],  waq1=P.w[G_AQ][wi1];
  const float wh0 =P.w[G_H][wi0],   wh1 =P.w[G_H][wi1];
  const float we0 =P.w[G_E][wi0],   we1 =P.w[G_E][wi1];
  const float wgg0=P.w[G_G][wi0],   wgg1=P.w[G_G][wi1];

  // combined biases
  const float bs0 = P.bw[G_SH][pi0]+P.bw[G_SX][pi0]+P.b_s[pi0];
  const float bs1 = P.bw[G_SH][pi1]+P.bw[G_SX][pi1]+P.b_s[pi1];
  const float bf0 = P.bw[G_FH][pi0]+P.bw[G_FX][pi0]+P.bw[G_FS][pi0]+P.b_f[pi0];
  const float bf1 = P.bw[G_FH][pi1]+P.bw[G_FX][pi1]+P.bw[G_FS][pi1]+P.b_f[pi1];
  const float bi0 = P.bw[G_IH][pi0]+P.bw[G_IX][pi0]+P.bw[G_IS][pi0]+P.b_i[pi0];
  const float bi1 = P.bw[G_IH][pi1]+P.bw[G_IX][pi1]+P.bw[G_IS][pi1]+P.b_i[pi1];
  const float bT0 = P.bw[G_TH][pi0]+P.bw[G_TX][pi0]+P.bw[G_TS][pi0]+P.b_T[pi0];
  const float bT1 = P.bw[G_TH][pi1]+P.bw[G_TX][pi1]+P.bw[G_TS][pi1]+P.b_T[pi1];
  const float bel0= P.bw[G_ELPH][pi0]+P.bw[G_ELPX][pi0]+P.bw[G_ELPS][pi0]+P.b_elp[wi0];
  const float bel1= P.bw[G_ELPH][pi1]+P.bw[G_ELPX][pi1]+P.bw[G_ELPS][pi1]+P.b_elp[wi1];
  const float bo0 = P.bw[G_OH][pi0]+P.bw[G_OX][pi0]+P.bw[G_OS][pi0]+P.b_o[pi0];
  const float bo1 = P.bw[G_OH][pi1]+P.bw[G_OX][pi1]+P.bw[G_OS][pi1]+P.b_o[pi1];
  const float baq0= P.bw[G_AQ][pi0]+P.bw[G_AH][pi0]+P.b_a[pi0];
  const float baq1= P.bw[G_AQ][pi1]+P.bw[G_AH][pi1]+P.b_a[pi1];
  const float bh0 = P.bw[G_H][pi0]+P.bw[G_E][pi0]+P.bw[G_G][pi0]+P.b_h[pi0];
  const float bh1 = P.bw[G_H][pi1]+P.bw[G_E][pi1]+P.bw[G_G][pi1]+P.b_h[pi1];

  const float wst  = P.w_st[wid];
  const float bwst = P.bw_st[wid];
  const float gm0 = P.gamma[d0], gm1 = P.gamma[d1];
  const float bt0 = P.beta[d0],  bt1 = P.beta[d1];

  float c_0 = 0.f, c_1 = 0.f;
  float h_0 = 0.f, h_1 = 0.f, hsum_last = 0.f;

  __syncthreads();

  for (int t = 0; t < UNITS; ++t) {
    const float hsum = memsumS[t * MSPAD + wid];          // sum_d h_t
    const float xsum = rowsumS[(t + kk) * PSZ + ps];      // sum_f x
    const int   tb   = (t + kk) * PSZ;
    const float tsumk = t0S[tb] + t0S[tb+1] + t0S[tb+2] + t0S[tb+3];
    const float tp    = tsumk * wst + bwst;

    const float s0 = sigm(hsum*wsh0 + xsum*wsx0 + tp + bs0);
    const float s1 = sigm(hsum*wsh1 + xsum*wsx1 + tp + bs1);
    const float ssum = wsum(s0 + s1);

    const float fg0 = sigm(hsum*wfh0 + xsum*wfx0 + ssum*wfs0 + bf0);
    const float fg1 = sigm(hsum*wfh1 + xsum*wfx1 + ssum*wfs1 + bf1);
    const float ig0 = sigm(hsum*wih0 + xsum*wix0 + ssum*wis0 + bi0);
    const float ig1 = sigm(hsum*wih1 + xsum*wix1 + ssum*wis1 + bi1);
    const float Tg0 = sigm(hsum*wTh0 + xsum*wTx0 + ssum*wTs0 + bT0);
    const float Tg1 = sigm(hsum*wTh1 + xsum*wTx1 + ssum*wTs1 + bT1);
    const float ch0 = fast_tanh(hsum*weh0 + xsum*wex0 + ssum*wes0 + bel0);
    const float ch1 = fast_tanh(hsum*weh1 + xsum*wex1 + ssum*wes1 + bel1);
    const float og0 = sigm(hsum*woh0 + xsum*wox0 + ssum*wos0 + bo0);
    const float og1 = sigm(hsum*woh1 + xsum*wox1 + ssum*wos1 + bo1);

    c_0 = fg0*c_0 + ig0*ch0 + Tg0*s0;
    c_1 = fg1*c_1 + ig1*ch1 + Tg1*s1;
    const float ht0 = og0 * fast_tanh(c_0);
    const float ht1 = og1 * fast_tanh(c_1);
    const float htsum = wsum(ht0 + ht1);

    // A_d = qa + bw_ah + b_a  (per-step attention bias vector)
    AbufS[wi0] = htsum*waq0 + baq0;
    AbufS[wi1] = htsum*waq1 + baq1;

    // usum partials, m-parallel across lanes (m>t rows zeroed)
    {
      const float* Aw = &AbufS[wid * DDIM];
      const float* Ww = &wahS[wid * DDIM];
      #pragma unroll
      for (int m0 = 0; m0 < UNITS; m0 += 32) {
        int m = m0 + lane;
        if (m < UNITS) {
          float acc = 0.f;
          if (m <= t) {
            const float ms = memsumS[m * MSPAD + wid];
            #pragma unroll 4
            for (int d = 0; d < DDIM; ++d)
              acc += fast_tanh(fmaf(ms, Ww[d], Aw[d]));
            acc *= wv;
          }
          partS[wid * UNITS + m] = acc;
        }
      }
    }
    __syncthreads();

    if (wid == 0) {
      // sc[m] = sum_{w=0..31} partial[w][m]  via  V_WMMA_F32_16X16X4_F32
      // A = ones(16x4), B = 4x16 chunk of partials, accumulate 8 w-chunks.
      const int nn = lane & 15;
      const int wr = (lane >> 4) << 1;
      v2f amat; amat[0] = 1.f; amat[1] = 1.f;
      #pragma unroll
      for (int mb = 0; mb < UNITS; mb += 16) {
        v8f cacc = {};
        #pragma unroll
        for (int wb = 0; wb < NW; wb += 4) {
          v2f bmat;
          bmat[0] = partS[(wb + wr)     * UNITS + mb + nn];
          bmat[1] = partS[(wb + wr + 1) * UNITS + mb + nn];
          cacc = __builtin_amdgcn_wmma_f32_16x16x4_f32(
              false, amat, false, bmat, (short)0, cacc, false, false);
        }
        if (lane < 16) scS[mb + lane] = cacc[0];
      }
      // softmax over valid m (m <= t)
      float v0m = (lane      <= t) ? scS[lane]      : -1e30f;
      float v1m = (lane + 32 <= t) ? scS[lane + 32] : -1e30f;
      float mx = wmaxr(fmaxf(v0m, v1m));
      float e0 = __expf(v0m - mx);
      float e1 = __expf(v1m - mx);
      float Z  = wsum(e0 + e1);
      float iz = 1.f / Z;
      alphaS[lane] = e0 * iz;
      if (lane < 16) alphaS[lane + 32] = e1 * iz;
    }
    __syncthreads();

    // attsum[k,ps] = sum_m alpha[m] * memsum[m,k,ps]
    float a0 = alphaS[lane] * memsumS[lane * MSPAD + wid];
    float a1 = (lane < 16) ? alphaS[lane + 32] * memsumS[(lane + 32) * MSPAD + wid] : 0.f;
    const float attsum = wsum(a0 + a1);

    h_0 = fast_tanh(htsum*wh0 + attsum*we0 + xsum*wgg0 + bh0);
    h_1 = fast_tanh(htsum*wh1 + attsum*we1 + xsum*wgg1 + bh1);
    const float hns = wsum(h_0 + h_1);
    hsum_last = hns;
    if (lane == 0) memsumS[(t + 1) * MSPAD + wid] = hns;
    __syncthreads();
  }

  // layernorm over D of final h, then write (B,K,PS,D)
  const float mean = hsum_last * (1.f / 64.f);
  const float dv0 = h_0 - mean, dv1 = h_1 - mean;
  const float var = wsum(dv0*dv0 + dv1*dv1) * (1.f / 64.f);
  const float rstd = rsqrtf(var + 1e-3f);
  float* ob = out + ((size_t)b * NW + wid) * DDIM;
  ob[d0] = dv0 * rstd * gm0 + bt0;
  ob[d1] = dv1 * rstd * gm1 + bt1;
}

extern "C" void kernel_launch(void* const* d_in, const int* in_sizes, int n_in,
                              void* d_out, int out_size, void* d_ws, size_t ws_size,
                              hipStream_t stream) {
  (void)in_sizes; (void)n_in; (void)out_size; (void)d_ws; (void)ws_size;
  Params P;
  const float* data = (const float*)d_in[0];
  for (int g = 0; g < NGATES; ++g) {
    P.w[g]  = (const float*)d_in[1 + 2*g];
    P.bw[g] = (const float*)d_in[2 + 2*g];
  }
  P.w_st  = (const float*)d_in[47];
  P.bw_st = (const float*)d_in[48];
  P.b_s   = (const float*)d_in[49];
  P.b_f   = (const float*)d_in[50];
  P.b_i   = (const float*)d_in[51];
  P.b_T   = (const float*)d_in[52];
  P.b_o   = (const float*)d_in[53];
  P.b_a   = (const float*)d_in[54];
  P.b_h   = (const float*)d_in[55];
  P.b_elp = (const float*)d_in[56];
  P.gamma = (const float*)d_in[57];
  P.beta  = (const float*)d_in[58];
  th_lstm_kernel<<<dim3(BATCH), dim3(1024), 0, stream>>>(data, P, (float*)d_out);
}